// DEMONetWeightGraph_3083786518800
// MI455X (gfx1250) — compile-verified
//
#include <hip/hip_runtime.h>
#include <hip/hip_bf16.h>

typedef __attribute__((ext_vector_type(2))) float v2f;
typedef __attribute__((ext_vector_type(8))) float v8f;

#define N_NODES   50000
#define N_EDGES   800000
#define IN_DIM    128
#define HIDDEN    256
#define N_CLASSES 10
#define N_GRAPHS  64
#define LDS_PAD   2

__device__ __forceinline__ float elu1(float v) {
    return v > 0.0f ? v : (__expf(v) - 1.0f);
}

// out[idx[i]] += 1  (deg = segment_sum(ones, src); cnt = segment_sum(ones, batch))
__global__ void accum_ones_kernel(const long long* __restrict__ idx,
                                  float* __restrict__ out, int n) {
    int i = blockIdx.x * blockDim.x + threadIdx.x;
    if (i < n) atomicAdd(&out[(int)idx[i]], 1.0f);
}

// S[src[e], :] += X[dst[e], :]   (D = 4 << dShift floats per row; one float4 per thread)
__global__ void edge_scatter_kernel(const long long* __restrict__ src,
                                    const long long* __restrict__ dst,
                                    const float* __restrict__ X,
                                    float* __restrict__ S,
                                    int dShift) {
    long long tid = (long long)blockIdx.x * blockDim.x + threadIdx.x;
    long long total = (long long)N_EDGES << dShift;
    if (tid >= total) return;
    int e  = (int)(tid >> dShift);
    int c4 = ((int)tid & ((1 << dShift) - 1)) << 2;
    int D  = 4 << dShift;
    int s = (int)src[e];
    int d = (int)dst[e];
    const float4 v = *(const float4*)(X + (size_t)d * D + c4);
    float* p = S + (size_t)s * D + c4;
    atomicAdd(p + 0, v.x);
    atomicAdd(p + 1, v.y);
    atomicAdd(p + 2, v.z);
    atomicAdd(p + 3, v.w);
}

// NM = NM / max(deg,1);  XM = deg>0 ? X : 0   (dShift = log2(D))
__global__ void mask_mean_kernel(const float* __restrict__ X,
                                 const float* __restrict__ deg,
                                 float* __restrict__ NM,
                                 float* __restrict__ XM,
                                 int dShift) {
    long long idx = (long long)blockIdx.x * blockDim.x + threadIdx.x;
    long long total = (long long)N_NODES << dShift;
    if (idx >= total) return;
    int i = (int)(idx >> dShift);
    float dg  = deg[i];
    float inv = 1.0f / fmaxf(dg, 1.0f);
    NM[idx] = NM[idx] * inv;
    XM[idx] = (dg > 0.0f) ? X[idx] : 0.0f;
}

// H[M x 256] = elu(A0@W0 + A1@W1 + A2@W2 + bias), all f32 via V_WMMA_F32_16X16X4_F32.
// One 512-thread block (16 waves) owns one 16-row M panel and all 16 N-tiles:
//   - the contiguous 16xK A panel is staged into LDS once (coalesced), then each
//     wave reads its A fragments via ds_load_b64 (row stride K+2 words -> no
//     64-bank conflicts).
//   - W loads are lane-coalesced and L2-resident (<= 768 KB total weights).
// M=50000, N=256 are multiples of 16 -> no guards, EXEC stays all-ones for WMMA.
__global__ void __launch_bounds__(512)
gemm3_bias_elu_kernel(const float* __restrict__ A0, const float* __restrict__ W0,
                      const float* __restrict__ A1, const float* __restrict__ W1,
                      const float* __restrict__ A2, const float* __restrict__ W2,
                      int K, int kShift, const float* __restrict__ bias,
                      float* __restrict__ H) {
    extern __shared__ float lds[];            // 16 * (K + LDS_PAD) floats

    const int lane = threadIdx.x & 31;
    const int wave = threadIdx.x >> 5;        // 0..15 == N-tile index
    const int half = lane >> 4;               // 0: lanes 0-15, 1: lanes 16-31
    const int mr   = lane & 15;               // A row within tile
    const int nl   = lane & 15;               // B/C/D column within tile
    const int m0   = blockIdx.x * 16;
    const int n0   = wave * 16;
    const int ldsStride = K + LDS_PAD;

    v8f acc = {};

    const float* As[3] = {A0, A1, A2};
    const float* Ws[3] = {W0, W1, W2};
#pragma unroll
    for (int p = 0; p < 3; ++p) {
        // Stage contiguous 16xK A panel into LDS (16*K is a multiple of 512).
        const float* Ablk = As[p] + (size_t)m0 * K;
        for (int t = threadIdx.x; t < 16 * K; t += 512) {
            int r = t >> kShift;
            int c = t & (K - 1);
            lds[r * ldsStride + c] = Ablk[t];
        }
        __syncthreads();

        // A lane layout (16x4 f32): a[v] = A[mr][k + 2*half + v]  (from LDS)
        const float* arow = lds + mr * ldsStride + 2 * half;
        // B lane layout (4x16 f32): b[v] = W[k + 2*half + v][n0 + nl]
        const float* wcol = Ws[p] + (size_t)(2 * half) * HIDDEN + n0 + nl;
#pragma unroll 8
        for (int k = 0; k < K; k += 4) {
            v2f a = *(const v2f*)(arow + k);   // ds_load_b64, 8B aligned
            v2f b;
            b.x = wcol[(size_t)k * HIDDEN];
            b.y = wcol[(size_t)(k + 1) * HIDDEN];
            acc = __builtin_amdgcn_wmma_f32_16x16x4_f32(
                /*neg_a=*/false, a, /*neg_b=*/false, b,
                /*c_mod=*/(short)0, acc, /*reuse_a=*/false, /*reuse_b=*/false);
        }
        __syncthreads();   // before next p overwrites the LDS panel
    }

    // C/D layout: VGPR r -> row m0 + r + 8*half, col n0 + nl
    const float bb = bias[n0 + nl];
    float* out0 = H + (size_t)(m0 + half * 8) * HIDDEN + n0 + nl;
#pragma unroll
    for (int r = 0; r < 8; ++r) {
        out0[(size_t)r * HIDDEN] = elu1(acc[r] + bb);
    }
}

// G[batch[i], c] += H[i, c]
__global__ void pool_sum_kernel(const float* __restrict__ H,
                                const long long* __restrict__ batch,
                                float* __restrict__ G) {
    long long idx = (long long)blockIdx.x * blockDim.x + threadIdx.x;
    long long total = (long long)N_NODES * HIDDEN;
    if (idx >= total) return;
    int i = (int)(idx >> 8);
    int c = (int)idx & 255;
    atomicAdd(&G[(int)batch[i] * HIDDEN + c], H[idx]);
}

// out[g, c] = (G[g,:]/max(cnt[g],1)) @ Wc[:, c] + bc[c]
__global__ void classifier_kernel(const float* __restrict__ G,
                                  const float* __restrict__ cnt,
                                  const float* __restrict__ Wc,
                                  const float* __restrict__ bc,
                                  float* __restrict__ out) {
    int g = blockIdx.x;
    int c = threadIdx.x;
    if (c >= N_CLASSES) return;
    float inv = 1.0f / fmaxf(cnt[g], 1.0f);
    float s = bc[c];
    const float* grow = G + (size_t)g * HIDDEN;
    for (int k = 0; k < HIDDEN; ++k)
        s = fmaf(grow[k] * inv, Wc[(size_t)k * N_CLASSES + c], s);
    out[g * N_CLASSES + c] = s;
}

extern "C" void kernel_launch(void* const* d_in, const int* in_sizes, int n_in,
                              void* d_out, int out_size, void* d_ws, size_t ws_size,
                              hipStream_t stream) {
    const float*     x     = (const float*)d_in[0];
    const long long* edge  = (const long long*)d_in[1];   // int64 (2, N_EDGES)
    const long long* src   = edge;
    const long long* dst   = edge + N_EDGES;
    const long long* batch = (const long long*)d_in[2];   // int64 (N_NODES)
    const float* Wg0 = (const float*)d_in[3];
    const float* Wl0 = (const float*)d_in[4];
    const float* Ws0 = (const float*)d_in[5];
    const float* b0  = (const float*)d_in[6];
    const float* Wg1 = (const float*)d_in[7];
    const float* Wl1 = (const float*)d_in[8];
    const float* Ws1 = (const float*)d_in[9];
    const float* b1  = (const float*)d_in[10];
    const float* Wc  = (const float*)d_in[11];
    const float* bc  = (const float*)d_in[12];
    float* out = (float*)d_out;

    // Workspace layout (floats)
    float* ws    = (float*)d_ws;
    float* deg   = ws + 0;                               // 50000
    float* cnt   = ws + 50048;                           // 64
    float* gpool = ws + 50112;                           // 64*256
    float* NM    = ws + 66560;                           // 50000*256 (neigh sum/mean)
    float* AM    = NM + (size_t)N_NODES * HIDDEN;        // 50000*256 (masked input)
    float* H     = AM + (size_t)N_NODES * HIDDEN;        // 50000*256
    float* H2    = H  + (size_t)N_NODES * HIDDEN;        // 50000*256

    // Zero accumulators: deg/cnt/gpool header + NM in one contiguous memset
    hipMemsetAsync(d_ws, 0, (66560 + (size_t)N_NODES * HIDDEN) * sizeof(float), stream);

    // degrees (per src-node edge count) and per-graph node counts
    accum_ones_kernel<<<(N_EDGES + 255) / 256, 256, 0, stream>>>(src, deg, N_EDGES);
    accum_ones_kernel<<<(N_NODES + 255) / 256, 256, 0, stream>>>(batch, cnt, N_NODES);

    // ---- layer 0 (D = 128) ----
    {
        long long total = (long long)N_EDGES * (IN_DIM / 4);
        edge_scatter_kernel<<<(int)((total + 255) / 256), 256, 0, stream>>>(src, dst, x, NM, 5);
        long long nd = (long long)N_NODES * IN_DIM;
        mask_mean_kernel<<<(int)((nd + 255) / 256), 256, 0, stream>>>(x, deg, NM, AM, 7);
    }
    dim3 ggrid(N_NODES / 16);   // 3125 M-panels; 16 waves/block cover all 16 N-tiles
    {
        size_t smem = 16 * (IN_DIM + LDS_PAD) * sizeof(float);
        gemm3_bias_elu_kernel<<<ggrid, 512, smem, stream>>>(x, Wg0, AM, Ws0, NM, Wl0,
                                                            IN_DIM, 7, b0, H);
    }

    // ---- layer 1 (D = 256) ----
    hipMemsetAsync(NM, 0, (size_t)N_NODES * HIDDEN * sizeof(float), stream);
    {
        long long total = (long long)N_EDGES * (HIDDEN / 4);
        edge_scatter_kernel<<<(int)((total + 255) / 256), 256, 0, stream>>>(src, dst, H, NM, 6);
        long long nd = (long long)N_NODES * HIDDEN;
        mask_mean_kernel<<<(int)((nd + 255) / 256), 256, 0, stream>>>(H, deg, NM, AM, 8);
    }
    {
        size_t smem = 16 * (HIDDEN + LDS_PAD) * sizeof(float);
        gemm3_bias_elu_kernel<<<ggrid, 512, smem, stream>>>(H, Wg1, AM, Ws1, NM, Wl1,
                                                            HIDDEN, 8, b1, H2);
    }

    // ---- pool + classifier ----
    {
        long long total = (long long)N_NODES * HIDDEN;
        pool_sum_kernel<<<(int)((total + 255) / 256), 256, 0, stream>>>(H2, batch, gpool);
    }
    classifier_kernel<<<N_GRAPHS, 32, 0, stream>>>(gpool, cnt, Wc, bc, out);
}